// AdditiveAttention_25546465477033
// MI455X (gfx1250) — compile-verified
//
#include <hip/hip_runtime.h>
#include <hip/hip_bf16.h>

#define NV    4096
#define NE    32768
#define DV    256
#define DATTN 128

typedef __attribute__((ext_vector_type(2))) float v2f;
typedef __attribute__((ext_vector_type(4))) float v4f;
typedef __attribute__((ext_vector_type(8))) float v8f;

// ---- order-preserving float<->int map for atomic max over f32 ----
__device__ __forceinline__ int orderF(float f) {
    int i = __float_as_int(f);
    return (i >= 0) ? i : (i ^ 0x7FFFFFFF);
}
__device__ __forceinline__ float unorderF(int i) {
    return __int_as_float((i >= 0) ? i : (i ^ 0x7FFFFFFF));
}

// ---- 1) zero-fill 512MB output with non-temporal b128 stores ----
__global__ void k_zero(v4f* __restrict__ out, size_t n4) {
    size_t i      = (size_t)blockIdx.x * blockDim.x + threadIdx.x;
    size_t stride = (size_t)gridDim.x * blockDim.x;
    v4f z = {0.f, 0.f, 0.f, 0.f};
    for (; i < n4; i += stride)
        __builtin_nontemporal_store(z, &out[i]);
}

// ---- 2) init per-node softmax state ----
__global__ void k_init(int* __restrict__ segmax, float* __restrict__ denom) {
    int i = blockIdx.x * blockDim.x + threadIdx.x;
    if (i < NV) {
        segmax[i] = orderF(-__builtin_inff());
        denom[i]  = 0.f;
    }
}

// ---- 3) u1 = Z^T w[:128], u2 = Z^T w[128:]  (256 threads, coalesced cols) ----
__global__ void k_u(const float* __restrict__ Z, const float* __restrict__ w,
                    float* __restrict__ u) {
    int d = threadIdx.x;  // 0..255
    float u1 = 0.f, u2 = 0.f;
    for (int k = 0; k < DATTN; ++k) {
        float z = Z[k * DV + d];
        u1 = fmaf(z, w[k], u1);
        u2 = fmaf(z, w[DATTN + k], u2);
    }
    u[d]      = u1;
    u[DV + d] = u2;
}

// ---- 4) (a,b) per node via V_WMMA_F32_16X16X4_F32 ----
// Each wave computes TWO 16x16 tiles: C{0,1}(16x16) = nodes_tile{0,1}(16x256) @ U(256x16),
// U cols 0/1 = u1/u2, rest zero. U is staged in LDS with a 2-float zero pad so that
// padding lanes (n>=2) do an unconditional ds_load_b64 from the zero slot (no EXEC
// divergence in the hot loop). Two independent accumulator chains hide WMMA RAW latency.
// A 16x4 f32 layout: lanes 0-15 hold M=0..15 {K=0,1}; lanes 16-31 {K=2,3}.
// B 4x16 f32 layout: lanes 0-15 hold N=0..15 {K=0,1}; lanes 16-31 {K=2,3}.
// D 16x16 f32: vgpr r = row r (lanes 0-15) / row 8+r (lanes 16-31), col = lane&15.
__global__ void __launch_bounds__(32) k_ab(const float* __restrict__ nodes,
                                           const float* __restrict__ u,
                                           float* __restrict__ ab) {
    __shared__ __align__(16) float lu[2 * DV + 2];  // u1 | u2 | {0,0} pad

    const int lane = threadIdx.x;
    // cooperative stage of u into LDS (512 floats / 32 lanes = 16 each)
    for (int i = lane; i < 2 * DV; i += 32) lu[i] = u[i];
    if (lane < 2) lu[2 * DV + lane] = 0.f;
    __syncthreads();

    const int half = lane >> 4;   // 0 or 1 -> K offset 0 or 2
    const int mn   = lane & 15;   // A row / B col
    const int tile = blockIdx.x;  // 32 nodes per wave (two 16-row tiles)

    const float* arow0 = nodes + ((size_t)tile * 32 + mn) * DV;
    const float* arow1 = arow0 + 16 * (size_t)DV;
    // B address: real u column for n<2, zero-pad slot for n>=2 (step 0 keeps it pinned)
    const float* bptr  = &lu[(mn < 2) ? mn * DV : 2 * DV];
    const int    bstep = (mn < 2) ? 1 : 0;

    v8f c0 = {0.f, 0.f, 0.f, 0.f, 0.f, 0.f, 0.f, 0.f};
    v8f c1 = {0.f, 0.f, 0.f, 0.f, 0.f, 0.f, 0.f, 0.f};
    for (int k0 = 0; k0 < DV; k0 += 4) {
        const int ka = k0 + half * 2;
        v2f A0 = *(const v2f*)(arow0 + ka);       // global_load_b64
        v2f A1 = *(const v2f*)(arow1 + ka);       // global_load_b64
        v2f B  = *(const v2f*)(bptr + ka * bstep);  // ds_load_b64 (zero slot for n>=2)
        c0 = __builtin_amdgcn_wmma_f32_16x16x4_f32(false, A0, false, B,
                                                   (short)0, c0, false, false);
        c1 = __builtin_amdgcn_wmma_f32_16x16x4_f32(false, A1, false, B,
                                                   (short)0, c1, false, false);
    }
    if (mn < 2) {  // only columns 0 (=a) and 1 (=b) are meaningful
        const int rbase = half * 8;
        for (int r = 0; r < 8; ++r) {
            ab[2 * ((size_t)tile * 32 + rbase + r) + mn]      = c0[r];
            ab[2 * ((size_t)tile * 32 + 16 + rbase + r) + mn] = c1[r];
        }
    }
}

// ---- 5) edge score + leaky relu + segment max ----
__global__ void k_pre(const int* __restrict__ ei, const float* __restrict__ ab,
                      float* __restrict__ pre, int* __restrict__ segmax) {
    int e = blockIdx.x * blockDim.x + threadIdx.x;
    if (e >= NE) return;
    int s = ei[e];
    int r = ei[NE + e];
    float x = ab[2 * s] + ab[2 * r + 1];
    x = (x > 0.f) ? x : 0.01f * x;  // LeakyReLU(0.01)
    pre[e] = x;
    atomicMax(&segmax[r], orderF(x));
}

// ---- 6) exp(pre - max) + segment sum ----
__global__ void k_ex(const int* __restrict__ ei, const int* __restrict__ segmax,
                     float* __restrict__ pre, float* __restrict__ denom) {
    int e = blockIdx.x * blockDim.x + threadIdx.x;
    if (e >= NE) return;
    int r = ei[NE + e];
    float ex = __expf(pre[e] - unorderF(segmax[r]));
    pre[e] = ex;  // overwrite in place
    atomicAdd(&denom[r], ex);
}

// ---- 7) scatter normalized weights into dense output ----
__global__ void k_scatter(const int* __restrict__ ei, const float* __restrict__ ex,
                          const float* __restrict__ denom, float* __restrict__ out) {
    int e = blockIdx.x * blockDim.x + threadIdx.x;
    if (e >= NE) return;
    int r = ei[NE + e];
    out[(size_t)r * NE + e] = ex[e] / denom[r];
}

extern "C" void kernel_launch(void* const* d_in, const int* in_sizes, int n_in,
                              void* d_out, int out_size, void* d_ws, size_t ws_size,
                              hipStream_t stream) {
    // setup_inputs order: nodes, Z, w, edge_index
    const float* nodes = (const float*)d_in[0];
    const float* Z     = (const float*)d_in[1];
    const float* w     = (const float*)d_in[2];
    const int*   ei    = (const int*)d_in[3];
    float*       out   = (float*)d_out;

    // workspace layout (floats): u[512] | ab[8192] | pre[32768] | segmax[4096] | denom[4096]
    float* u      = (float*)d_ws;
    float* ab     = u + 2 * DV;
    float* pre    = ab + 2 * NV;
    int*   segmax = (int*)(pre + NE);
    float* denom  = (float*)(segmax + NV);

    const size_t n4 = (size_t)NV * NE / 4;  // float4 count for zero fill

    k_zero<<<32768, 256, 0, stream>>>((v4f*)out, n4);
    k_init<<<NV / 256, 256, 0, stream>>>(segmax, denom);
    k_u<<<1, 256, 0, stream>>>(Z, w, u);
    k_ab<<<NV / 32, 32, 0, stream>>>(nodes, u, ab);
    k_pre<<<NE / 256, 256, 0, stream>>>(ei, ab, pre, segmax);
    k_ex<<<NE / 256, 256, 0, stream>>>(ei, segmax, pre, denom);
    k_scatter<<<NE / 256, 256, 0, stream>>>(ei, pre, denom, out);
}